// ReadoutLayer_90847148245287
// MI455X (gfx1250) — compile-verified
//
#include <hip/hip_runtime.h>

typedef __attribute__((ext_vector_type(2))) float v2f;
typedef __attribute__((ext_vector_type(8))) float v8f;

#define GRAPHS 4096
#define NODES  64
#define DIM    256
#define SKIP_OP 5

__global__ __launch_bounds__(256) void readout_wmma_kernel(
    const float* __restrict__ h, const int* __restrict__ op, float* __restrict__ out)
{
    __shared__ float smask[NODES];
    __shared__ float sinv;

    const int b    = blockIdx.x;
    const int tid  = threadIdx.x;
    const int lane = tid & 31;
    const int wave = tid >> 5;

    // Build 0/1 mask for this graph's 64 nodes in LDS.
    if (tid < NODES) {
        smask[tid] = (op[b * NODES + tid] != SKIP_OP) ? 1.0f : 0.0f;
    }
    __syncthreads();
    if (tid == 0) {
        float cnt = 0.0f;
        #pragma unroll
        for (int i = 0; i < NODES; ++i) cnt += smask[i];
        sinv = 1.0f / fmaxf(cnt, 1.0f);
    }
    __syncthreads();

    const float inv  = sinv;
    const int   half = lane >> 4;    // 0 for lanes 0-15, 1 for lanes 16-31
    const int   col  = lane & 15;    // N / column-in-tile index
    const float* hbase = h + (size_t)b * NODES * DIM;

    // Each wave handles columns [32*wave, 32*wave+32): two 16-col WMMA tiles.
    #pragma unroll
    for (int t = 0; t < 2; ++t) {
        const int d0 = wave * 32 + t * 16;
        v8f c = {};

        // Reduce over all 64 nodes in K-chunks of 4 with f32 WMMA.
        #pragma unroll
        for (int ch = 0; ch < 16; ++ch) {
            const int k0 = ch * 4 + half * 2;   // this half-wave's two K rows

            // A fragment: mask broadcast into all 16 M rows.
            v2f a;
            a.x = smask[k0];
            a.y = smask[k0 + 1];

            // B fragment: h[node k0][d0+col], h[node k0+1][d0+col]
            const float* hp = hbase + (size_t)k0 * DIM + d0 + col;
            v2f bf;
            bf.x = hp[0];
            bf.y = hp[DIM];

            // D = A*B + C  (exact f32 path: v_wmma_f32_16x16x4_f32)
            c = __builtin_amdgcn_wmma_f32_16x16x4_f32(
                    /*neg_a=*/false, a, /*neg_b=*/false, bf,
                    /*c_mod=*/(short)0, c, /*reuse_a=*/false, /*reuse_b=*/false);
        }

        // Row M=0 of D: VGPR 0, lanes 0-15, N = lane. All rows identical.
        if (lane < 16) {
            out[(size_t)b * DIM + d0 + col] = c[0] * inv;
        }
    }
}

extern "C" void kernel_launch(void* const* d_in, const int* in_sizes, int n_in,
                              void* d_out, int out_size, void* d_ws, size_t ws_size,
                              hipStream_t stream) {
    const float* h  = (const float*)d_in[0];
    const int*   op = (const int*)d_in[1];
    float*       out = (float*)d_out;
    readout_wmma_kernel<<<GRAPHS, 256, 0, stream>>>(h, op, out);
}